// Predic_Net_39651138076737
// MI455X (gfx1250) — compile-verified
//
#include <hip/hip_runtime.h>
#include <hip/hip_bf16.h>
#include <math.h>

// Shapes fixed by setup_inputs(): B=32, n=1024, D=512, m=1023, gate rows=1022.
#define B_    32
#define N_    1024
#define D_    512
#define M_    1023
#define GROWS 1022
#define SQRT512_F 22.62741699796952f

typedef float v2f __attribute__((ext_vector_type(2)));
typedef float v8f __attribute__((ext_vector_type(8)));

// ---------------------------------------------------------------------------
// Kernel A: U = [rep[b,0,:] (32 rows) ; rep[b,1023,:] (32 rows)] @ W
//           (64 x 512) = (64 x 512) @ (512 x 512), fp32 WMMA 16x16x4.
// One wave (32 threads) per 16x16 output tile; 4 row-tiles * 32 col-tiles.
// ---------------------------------------------------------------------------
__global__ __launch_bounds__(32)
void k_u_gemm(const float* __restrict__ rep, const float* __restrict__ W,
              float* __restrict__ U) {
  const int lane = threadIdx.x;           // 0..31
  const int mt   = blockIdx.x >> 5;       // 0..3
  const int nt   = blockIdx.x & 31;       // 0..31
  const int m0 = mt << 4, n0 = nt << 4;
  const int half = lane >> 4;             // 0/1
  const int l15  = lane & 15;
  const int kb   = half << 1;             // K sub-offset: 0 or 2

  // A-matrix row for this lane (16x4 tile: lane = M row)
  const int grow = m0 + l15;              // global row 0..63
  const float* Arow = (grow < B_)
      ? rep + (size_t)grow * (N_ * D_)                       // rep[b,0,:]
      : rep + ((size_t)(grow - B_) * N_ + (N_ - 1)) * D_;    // rep[b,1023,:]
  const float* Bcol = W + n0 + l15;       // B tile: lane = N column

  v8f c = {};
  for (int k0 = 0; k0 < D_; k0 += 4) {
    v2f a, b;
    a.x = Arow[k0 + kb];
    a.y = Arow[k0 + kb + 1];
    b.x = Bcol[(size_t)(k0 + kb) * D_];
    b.y = Bcol[(size_t)(k0 + kb + 1) * D_];
    c = __builtin_amdgcn_wmma_f32_16x16x4_f32(
        /*neg_a=*/false, a, /*neg_b=*/false, b,
        /*c_mod=*/(short)0, c, /*reuse_a=*/false, /*reuse_b=*/false);
  }
  // C/D layout: VGPR r -> M = m0 + r + half*8, N = n0 + (lane&15)
#pragma unroll
  for (int r = 0; r < 8; ++r)
    U[(size_t)(m0 + r + half * 8) * D_ + n0 + l15] = c[r];
}

// ---------------------------------------------------------------------------
// Kernel B: s[b,j] = diag * colsum for both gates.
//   colsum = 1 + sum_{i<j} gate[b,i,j]; diag = (j==0) ? 1 : gate[b,j-1,j].
// Thread j walks column j (coalesced across threads within a row of gate).
// ---------------------------------------------------------------------------
__global__ __launch_bounds__(256)
void k_scales(const float* __restrict__ gf, const float* __restrict__ gb,
              float* __restrict__ sf, float* __restrict__ sb) {
  const int b = blockIdx.x >> 2;
  const int j = ((blockIdx.x & 3) << 8) + threadIdx.x;
  if (j >= M_) return;
  const float* pf = gf + (size_t)b * GROWS * M_ + j;
  const float* pb = gb + (size_t)b * GROWS * M_ + j;
  float cf = 1.f, cb = 1.f;
#pragma unroll 4
  for (int i = 0; i < j; ++i) {
    cf += pf[(size_t)i * M_];
    cb += pb[(size_t)i * M_];
  }
  const float df = (j == 0) ? 1.f : pf[(size_t)(j - 1) * M_];
  const float db = (j == 0) ? 1.f : pb[(size_t)(j - 1) * M_];
  sf[b * M_ + j] = df * cf;
  sb[b * M_ + j] = db * cb;
}

// ---------------------------------------------------------------------------
// Wave32 pair of 512-length dot products: d0 = t0.u0, d1 = t1.u1.
// float4 loads, fully coalesced; shuffle-xor reduction (deterministic).
// ---------------------------------------------------------------------------
__device__ __forceinline__ void dot_pair(const float* __restrict__ t0,
                                         const float* __restrict__ u0,
                                         const float* __restrict__ t1,
                                         const float* __restrict__ u1,
                                         int lane, float& d0, float& d1) {
  const float4* t04 = (const float4*)t0;
  const float4* u04 = (const float4*)u0;
  const float4* t14 = (const float4*)t1;
  const float4* u14 = (const float4*)u1;
  float a = 0.f, b = 0.f;
#pragma unroll
  for (int i = 0; i < 4; ++i) {
    const int idx = (i << 5) + lane;      // 512 floats = 128 float4
    float4 x = t04[idx], y = u04[idx];
    a = fmaf(x.x, y.x, a); a = fmaf(x.y, y.y, a);
    a = fmaf(x.z, y.z, a); a = fmaf(x.w, y.w, a);
    float4 z = t14[idx], w = u14[idx];
    b = fmaf(z.x, w.x, b); b = fmaf(z.y, w.y, b);
    b = fmaf(z.z, w.z, b); b = fmaf(z.w, w.w, b);
  }
#pragma unroll
  for (int off = 16; off > 0; off >>= 1) {
    a += __shfl_xor(a, off, 32);
    b += __shfl_xor(b, off, 32);
  }
  d0 = a; d1 = b;
}

__device__ __forceinline__ float log_sig(float x) {
  // numerically stable log(sigmoid(x)) = min(x,0) - log1p(exp(-|x|))
  return fminf(x, 0.f) - log1pf(expf(-fabsf(x)));
}

// ---------------------------------------------------------------------------
// Kernel C (positives): wave per rep row p = b*1024 + t. Row is read once and
// feeds BOTH fwd-pos (j = t-1) and bwd-pos (j = 1022-t) logits.
// 4096 blocks * 8 waves = 32768 rows exactly.
// ---------------------------------------------------------------------------
__global__ __launch_bounds__(256)
void k_pos(const float* __restrict__ rep, const float* __restrict__ U,
           const float* __restrict__ sf, const float* __restrict__ sb,
           const float* __restrict__ bptr, float* __restrict__ partials) {
  __shared__ float sm[8];
  const int tid = threadIdx.x, lane = tid & 31, wid = tid >> 5;
  const int p = blockIdx.x * 8 + wid;     // 0..32767
  const int b = p >> 10, t = p & (N_ - 1);
  const float* trow = rep + (size_t)p * D_;
  const float* uf = U + (size_t)b * D_;
  const float* ub = U + (size_t)(b + B_) * D_;
  float df, db;
  dot_pair(trow, uf, trow, ub, lane, df, db);
  float contrib = 0.f;
  if (lane == 0) {
    const float b0 = bptr[0];
    float acc = 0.f;
    if (t >= 1)                           // fwd positive, j = t-1
      acc += log_sig((sf[b * M_ + (t - 1)] * df + b0) / SQRT512_F);
    if (t <= N_ - 2)                      // bwd positive, j = 1022-t
      acc += log_sig((sb[b * M_ + (N_ - 2 - t)] * db + b0) / SQRT512_F);
    contrib = 0.5f * acc;
  }
  if (lane == 0) sm[wid] = contrib;
  __syncthreads();
  if (tid == 0) {
    float s = 0.f;
#pragma unroll
    for (int i = 0; i < 8; ++i) s += sm[i];
    partials[blockIdx.x] = s;
  }
}

// ---------------------------------------------------------------------------
// Kernel D (negatives): wave per q = b*1023 + j. 4092 blocks * 8 = 32736.
// ---------------------------------------------------------------------------
__global__ __launch_bounds__(256)
void k_neg(const float* __restrict__ fneg, const float* __restrict__ bneg,
           const float* __restrict__ U, const float* __restrict__ sf,
           const float* __restrict__ sb, const float* __restrict__ bptr,
           float* __restrict__ partials) {
  __shared__ float sm[8];
  const int tid = threadIdx.x, lane = tid & 31, wid = tid >> 5;
  const int q = blockIdx.x * 8 + wid;     // 0..32735
  const int b = q / M_;
  const int j = q - b * M_;
  const float* fr = fneg + (size_t)q * D_;
  const float* br = bneg + (size_t)q * D_;
  const float* uf = U + (size_t)b * D_;
  const float* ub = U + (size_t)(b + B_) * D_;
  float dfn, dbn;
  dot_pair(fr, uf, br, ub, lane, dfn, dbn);
  float contrib = 0.f;
  if (lane == 0) {
    const float b0 = bptr[0];
    const float lf = -((sf[b * M_ + j] * dfn + b0) / SQRT512_F);
    const float lb = -((sb[b * M_ + j] * dbn + b0) / SQRT512_F);
    contrib = 0.5f * (log_sig(lf) + log_sig(lb));
  }
  if (lane == 0) sm[wid] = contrib;
  __syncthreads();
  if (tid == 0) {
    float s = 0.f;
#pragma unroll
    for (int i = 0; i < 8; ++i) s += sm[i];
    partials[blockIdx.x] = s;
  }
}

// ---------------------------------------------------------------------------
// Final deterministic reduction: fixed strided order + shared tree.
// ---------------------------------------------------------------------------
__global__ __launch_bounds__(256)
void k_final(const float* __restrict__ pp, int npos,
             const float* __restrict__ nn, int nneg,
             float* __restrict__ out, float invN) {
  __shared__ float sA[256], sB[256];
  const int tid = threadIdx.x;
  float a = 0.f, b = 0.f;
  for (int i = tid; i < npos; i += 256) a += pp[i];
  for (int i = tid; i < nneg; i += 256) b += nn[i];
  sA[tid] = a; sB[tid] = b;
  __syncthreads();
  for (int s = 128; s > 0; s >>= 1) {
    if (tid < s) { sA[tid] += sA[tid + s]; sB[tid] += sB[tid + s]; }
    __syncthreads();
  }
  if (tid == 0) { out[0] = sA[0] * invN; out[1] = sB[0] * invN; }
}

extern "C" void kernel_launch(void* const* d_in, const int* in_sizes, int n_in,
                              void* d_out, int out_size, void* d_ws, size_t ws_size,
                              hipStream_t stream) {
  (void)in_sizes; (void)n_in; (void)out_size; (void)ws_size;
  const float* rep  = (const float*)d_in[0];
  const float* gf   = (const float*)d_in[1];
  const float* gb   = (const float*)d_in[2];
  const float* fneg = (const float*)d_in[3];
  const float* bneg = (const float*)d_in[4];
  const float* W    = (const float*)d_in[5];
  const float* bb   = (const float*)d_in[6];
  float* out = (float*)d_out;

  // Workspace layout (floats): U[64*512] | sf[32736] pad | sb[32736] pad | partials
  float* ws = (float*)d_ws;
  float* U  = ws;             // 32768 floats
  float* sf = ws + 32768;     // 32736 floats (padded to 32768)
  float* sb = ws + 65536;     // 32736 floats (padded to 32768)
  float* pp = ws + 98304;     // 4096 pos partials
  float* np = ws + 102400;    // 4092 neg partials

  k_u_gemm<<<128, 32, 0, stream>>>(rep, W, U);
  k_scales<<<B_ * 4, 256, 0, stream>>>(gf, gb, sf, sb);
  k_pos<<<4096, 256, 0, stream>>>(rep, U, sf, sb, bb, pp);
  k_neg<<<4092, 256, 0, stream>>>(fneg, bneg, U, sf, sb, bb, np);
  k_final<<<1, 256, 0, stream>>>(pp, 4096, np, 4092, out, 1.0f / 32736.0f);
}